// PointNetFeaturePropagation_64596308132331
// MI455X (gfx1250) — compile-verified
//
#include <hip/hip_runtime.h>
#include <hip/hip_bf16.h>
#include <math.h>

typedef float v2f __attribute__((ext_vector_type(2)));
typedef float v8f __attribute__((ext_vector_type(8)));

#define BATCH   16
#define NPTS    4096
#define SPTS    1024
#define C1CH    128
#define C2CH    256
#define INCH    384
#define NCOL    65536           // B * N
#define TDIM    256
#define COLGRPS 1024            // NCOL / 64
#define JSTRIDE 160             // floats per K-pair row in LDS (bank-conflict-free)

__device__ __forceinline__ float gelu_exact(float x) {
    return 0.5f * x * (1.0f + erff(x * 0.70710678118654752440f));
}

// ---------------------------------------------------------------------------
// Kernel 1: 3-NN search + inverse-distance weights. One thread per (b, n).
// xyz2 for the batch is staged in LDS (1024 * 3 floats = 12 KB).
// ---------------------------------------------------------------------------
__global__ void knn_interp_weights(const float* __restrict__ xyz1,
                                   const float* __restrict__ xyz2,
                                   int* __restrict__ idxb,
                                   float* __restrict__ wgtb) {
    __shared__ float sx[SPTS], sy[SPTS], sz[SPTS];
    const int b = blockIdx.y;
    const int n = blockIdx.x * 256 + threadIdx.x;

    const float* p2 = xyz2 + (size_t)b * SPTS * 3;
    for (int i = threadIdx.x; i < SPTS; i += 256) {
        sx[i] = p2[i * 3 + 0];
        sy[i] = p2[i * 3 + 1];
        sz[i] = p2[i * 3 + 2];
    }
    __syncthreads();

    const float* p = xyz1 + ((size_t)b * NPTS + n) * 3;
    const float px = p[0], py = p[1], pz = p[2];

    float d0 = 3.402823466e+38f, d1 = d0, d2 = d0;
    int   i0 = 0, i1 = 0, i2 = 0;
    #pragma unroll 4
    for (int s = 0; s < SPTS; ++s) {
        const float dx = px - sx[s], dy = py - sy[s], dz = pz - sz[s];
        const float d = dx * dx + dy * dy + dz * dz;
        if (d < d0)      { d2 = d1; i2 = i1; d1 = d0; i1 = i0; d0 = d; i0 = s; }
        else if (d < d1) { d2 = d1; i2 = i1; d1 = d;  i1 = s; }
        else if (d < d2) { d2 = d;  i2 = s; }
    }
    const float r0 = 1.0f / (d0 + 1e-8f);
    const float r1 = 1.0f / (d1 + 1e-8f);
    const float r2 = 1.0f / (d2 + 1e-8f);
    const float rs = 1.0f / (r0 + r1 + r2);

    const size_t o = ((size_t)b * NPTS + n) * 3;
    idxb[o + 0] = i0; idxb[o + 1] = i1; idxb[o + 2] = i2;
    wgtb[o + 0] = r0 * rs; wgtb[o + 1] = r1 * rs; wgtb[o + 2] = r2 * rs;
}

// ---------------------------------------------------------------------------
// Kernel 2: build Xcat (384 rows x 65536 cols, row-major).
// Rows [0,128): copy of points1 ; rows [128,384): weighted gather of points2
// (points2 is 16 MB -> fully L2-resident on MI455X's 192 MB L2).
// ---------------------------------------------------------------------------
__global__ void build_xcat(const float* __restrict__ points1,
                           const float* __restrict__ points2,
                           const int* __restrict__ idxb,
                           const float* __restrict__ wgtb,
                           float* __restrict__ Xcat) {
    const int row = blockIdx.y;                      // 0..383
    const int col = blockIdx.x * 256 + threadIdx.x;  // 0..65535
    const int b = col >> 12;
    const int n = col & (NPTS - 1);

    float v;
    if (row < C1CH) {
        v = points1[((size_t)b * C1CH + row) * NPTS + n];
    } else {
        const int c2 = row - C1CH;
        const size_t o = (size_t)col * 3;
        const float* prow = points2 + ((size_t)b * C2CH + c2) * SPTS;
        v = wgtb[o + 0] * prow[idxb[o + 0]]
          + wgtb[o + 1] * prow[idxb[o + 1]]
          + wgtb[o + 2] * prow[idxb[o + 2]];
    }
    Xcat[(size_t)row * NCOL + col] = v;
}

// ---------------------------------------------------------------------------
// Kernel 3: FP32 WMMA GEMM  Y = W * X + bias  with per-(colgroup,channel)
// partial BN statistics (deterministic: each slot written by exactly one wave).
//   W: (Cout, Cin) row-major ; X: (Cin, 65536) row-major ; Y: (Cout, 65536).
//
// The 64-column X strip (CIN x 64 fp32) is staged ONCE per workgroup in LDS,
// stored K-pair-interleaved: sX[j*JSTRIDE + 2*c] = {X[2j][c], X[2j+1][c]},
// so a WMMA B-fragment is a single contiguous float2 -> one ds_load_b64.
// JSTRIDE=160 floats => lane-half row offset = 160 floats = 32 banks (mod 64):
// lanes 0-15 hit banks 0-31 while lanes 16-31 hit banks 32-63 (conflict-free).
//
// V_WMMA_F32_16X16X4_F32 fragment layout (ISA 7.12.2):
//   A 16x4 : lanes 0-15 hold M=lane; VGPR0 = K {0 | 2}, VGPR1 = K {1 | 3}
//            selected by lane-half.
//   C/D    : VGPR r, lanes 0-15 -> row r, lanes 16-31 -> row r+8.
// ---------------------------------------------------------------------------
template <int CIN>
__global__ void gemm_bn(const float* __restrict__ X,
                        const float* __restrict__ W,
                        const float* __restrict__ bias,
                        float* __restrict__ Y,
                        float* __restrict__ psum,
                        float* __restrict__ psq,
                        int Cout) {
    __shared__ float sX[(CIN / 2) * JSTRIDE];        // 384: 120 KB ; 256: 80 KB

    const int lane  = threadIdx.x & 31;
    const int wave  = threadIdx.x >> 5;
    const int l15   = lane & 15;
    const int khalf = lane >> 4;              // 0 or 1
    const int cb    = blockIdx.x * 64;        // column base of this block
    const int m0    = (blockIdx.y * 8 + wave) * 16;

    // ---- stage X strip into LDS (coalesced reads, interleaved K-pairs) ----
    for (int i = threadIdx.x; i < (CIN / 2) * 64; i += 256) {
        const int j = i >> 6;                 // K-pair index
        const int c = i & 63;                 // column within strip
        v2f p;
        p.x = X[(size_t)(2 * j + 0) * NCOL + cb + c];
        p.y = X[(size_t)(2 * j + 1) * NCOL + cb + c];
        *(v2f*)&sX[j * JSTRIDE + 2 * c] = p;
    }
    __syncthreads();
    if (m0 >= Cout) return;

    // Accumulators initialized with the conv bias (BN stats include bias).
    v8f acc[4];
    #pragma unroll
    for (int r = 0; r < 8; ++r) {
        const float bv = bias[m0 + r + khalf * 8];
        acc[0][r] = bv; acc[1][r] = bv; acc[2][r] = bv; acc[3][r] = bv;
    }

    const float* Wrow = W + (size_t)(m0 + l15) * CIN + (khalf << 1);
    for (int k = 0; k < CIN; k += 4) {
        __builtin_prefetch(Wrow + k + 64, 0, 0);      // global_prefetch_b8
        v2f a;
        a.x = Wrow[k];       // W[m0+l15][k + 2*khalf]
        a.y = Wrow[k + 1];   // W[m0+l15][k + 2*khalf + 1]

        const float* bbase = &sX[((k >> 1) + khalf) * JSTRIDE + 2 * l15];
        #pragma unroll
        for (int t = 0; t < 4; ++t) {
            const v2f bf = *(const v2f*)(bbase + 32 * t);   // ds_load_b64
            acc[t] = __builtin_amdgcn_wmma_f32_16x16x4_f32(
                false, a, false, bf, (short)0, acc[t], false, false);
        }
    }

    // Store Y: acc[t][r] at lane -> row m0 + r + khalf*8, col cb + 16*t + l15.
    #pragma unroll
    for (int t = 0; t < 4; ++t)
        #pragma unroll
        for (int r = 0; r < 8; ++r)
            Y[(size_t)(m0 + r + khalf * 8) * NCOL + cb + 16 * t + l15] = acc[t][r];

    // Partial BN stats over this wave's 64 columns, per its 16 channels.
    #pragma unroll
    for (int r = 0; r < 8; ++r) {
        float s = acc[0][r] + acc[1][r] + acc[2][r] + acc[3][r];
        float q = acc[0][r] * acc[0][r] + acc[1][r] * acc[1][r]
                + acc[2][r] * acc[2][r] + acc[3][r] * acc[3][r];
        #pragma unroll
        for (int off = 1; off < 16; off <<= 1) {
            s += __shfl_xor(s, off, 32);
            q += __shfl_xor(q, off, 32);
        }
        if (l15 == 0) {
            const int ch = m0 + r + khalf * 8;
            psum[(size_t)blockIdx.x * Cout + ch] = s;
            psq [(size_t)blockIdx.x * Cout + ch] = q;
        }
    }
}

// ---------------------------------------------------------------------------
// Kernel 4: fold partial stats + gamma/beta into per-channel scale/shift.
// ---------------------------------------------------------------------------
__global__ void finalize_stats(const float* __restrict__ psum,
                               const float* __restrict__ psq,
                               const float* __restrict__ gam,
                               const float* __restrict__ bet,
                               float* __restrict__ scale,
                               float* __restrict__ shift,
                               int Cout) {
    const int c = threadIdx.x;
    if (c >= Cout) return;
    float s = 0.0f, q = 0.0f;
    for (int g = 0; g < COLGRPS; ++g) {
        s += psum[(size_t)g * Cout + c];
        q += psq [(size_t)g * Cout + c];
    }
    const float mean = s * (1.0f / (float)NCOL);
    const float var  = q * (1.0f / (float)NCOL) - mean * mean;
    const float inv  = rsqrtf(var + 1e-5f);
    const float sc   = gam[c] * inv;
    scale[c] = sc;
    shift[c] = bet[c] - mean * sc;
}

// ---------------------------------------------------------------------------
// Kernel 5: te[b, o] = gelu(t_embed[b]) @ tw[o]^T + tb[o]   (tiny GEMM)
// ---------------------------------------------------------------------------
__global__ void temb_mlp(const float* __restrict__ t_embed,
                         const float* __restrict__ tw,
                         const float* __restrict__ tb,
                         float* __restrict__ te,
                         int Cout) {
    __shared__ float ge[TDIM];
    const int b = blockIdx.x;
    const int t = threadIdx.x;
    ge[t] = gelu_exact(t_embed[(size_t)b * TDIM + t]);
    __syncthreads();
    if (t < Cout) {
        float s = tb[t];
        const float* wr = tw + (size_t)t * TDIM;
        #pragma unroll 4
        for (int k = 0; k < TDIM; ++k) s += ge[k] * wr[k];
        te[(size_t)b * Cout + t] = s;
    }
}

// ---------------------------------------------------------------------------
// Kernel 6: elementwise BN-apply + t-embed bias + exact GELU.
// FINAL=false: write (Cout, 65536) row-major (next layer's X).
// FINAL=true : write output layout (B, 128, N).
// ---------------------------------------------------------------------------
template <bool FINAL>
__global__ void apply_act(const float* __restrict__ Y,
                          const float* __restrict__ scale,
                          const float* __restrict__ shift,
                          const float* __restrict__ te,
                          float* __restrict__ out,
                          int Cout) {
    const int c   = blockIdx.y;
    const int col = blockIdx.x * 256 + threadIdx.x;
    const int b   = col >> 12;
    float v = Y[(size_t)c * NCOL + col];
    v = v * scale[c] + shift[c] + te[(size_t)b * Cout + c];
    v = gelu_exact(v);
    if (FINAL)
        out[(size_t)b * (128 * NPTS) + (size_t)c * NPTS + (col & (NPTS - 1))] = v;
    else
        out[(size_t)c * NCOL + col] = v;
}

// ---------------------------------------------------------------------------
extern "C" void kernel_launch(void* const* d_in, const int* in_sizes, int n_in,
                              void* d_out, int out_size, void* d_ws, size_t ws_size,
                              hipStream_t stream) {
    (void)in_sizes; (void)n_in; (void)out_size; (void)ws_size;

    const float* xyz1    = (const float*)d_in[0];
    const float* xyz2    = (const float*)d_in[1];
    const float* points1 = (const float*)d_in[2];
    const float* points2 = (const float*)d_in[3];
    const float* t_embed = (const float*)d_in[4];
    const float* w0  = (const float*)d_in[5];
    const float* b0  = (const float*)d_in[6];
    const float* g0  = (const float*)d_in[7];
    const float* be0 = (const float*)d_in[8];
    const float* tw0 = (const float*)d_in[9];
    const float* tb0 = (const float*)d_in[10];
    const float* w1  = (const float*)d_in[11];
    const float* b1  = (const float*)d_in[12];
    const float* g1  = (const float*)d_in[13];
    const float* be1 = (const float*)d_in[14];
    const float* tw1 = (const float*)d_in[15];
    const float* tb1 = (const float*)d_in[16];
    float* out = (float*)d_out;

    // Workspace carve-up (256B aligned regions).
    char* w = (char*)d_ws;
    auto take = [&](size_t bytes) {
        char* p = w;
        w += (bytes + 255) & ~(size_t)255;
        return p;
    };
    int*   idxb  = (int*)  take((size_t)NCOL * 3 * sizeof(int));      // 0.75 MB
    float* wgtb  = (float*)take((size_t)NCOL * 3 * sizeof(float));    // 0.75 MB
    float* Xcat  = (float*)take((size_t)INCH * NCOL * sizeof(float)); // 100.7 MB
    float* Ybuf  = (float*)take((size_t)256  * NCOL * sizeof(float)); // 67.1 MB
    float* psum  = (float*)take((size_t)COLGRPS * 256 * sizeof(float)); // 1 MB
    float* psq   = (float*)take((size_t)COLGRPS * 256 * sizeof(float)); // 1 MB
    float* te    = (float*)take((size_t)BATCH * 256 * sizeof(float));
    float* scl   = (float*)take(256 * sizeof(float));
    float* shf   = (float*)take(256 * sizeof(float));
    float* X1    = Xcat;   // layer-1 input aliases Xcat (free after GEMM0)
    float* Y1    = Ybuf;   // layer-1 pre-activation aliases Y0 (free after apply0)

    // 1) 3-NN + weights
    knn_interp_weights<<<dim3(NPTS / 256, BATCH), 256, 0, stream>>>(xyz1, xyz2, idxb, wgtb);
    // 2) concat [points1 ; interp]
    build_xcat<<<dim3(NCOL / 256, INCH), 256, 0, stream>>>(points1, points2, idxb, wgtb, Xcat);

    // ---- layer 0: 384 -> 256 ----
    temb_mlp<<<BATCH, TDIM, 0, stream>>>(t_embed, tw0, tb0, te, 256);
    gemm_bn<INCH><<<dim3(COLGRPS, 2), 256, 0, stream>>>(Xcat, w0, b0, Ybuf, psum, psq, 256);
    finalize_stats<<<1, 256, 0, stream>>>(psum, psq, g0, be0, scl, shf, 256);
    apply_act<false><<<dim3(NCOL / 256, 256), 256, 0, stream>>>(Ybuf, scl, shf, te, X1, 256);

    // ---- layer 1: 256 -> 128 ----
    temb_mlp<<<BATCH, TDIM, 0, stream>>>(t_embed, tw1, tb1, te, 128);
    gemm_bn<256><<<dim3(COLGRPS, 1), 256, 0, stream>>>(X1, w1, b1, Y1, psum, psq, 128);
    finalize_stats<<<1, 256, 0, stream>>>(psum, psq, g1, be1, scl, shf, 128);
    apply_act<true><<<dim3(NCOL / 256, 128), 256, 0, stream>>>(Y1, scl, shf, te, out, 128);
}